// Plastic_80410377716450
// MI455X (gfx1250) — compile-verified
//
#include <hip/hip_runtime.h>
#include <hip/hip_bf16.h>

#define NF        512
#define T_STEPS   2048
#define G_BLOCKS  32
#define ROWS      (NF / G_BLOCKS)   // 16 rows of trace per workgroup

typedef __attribute__((ext_vector_type(2))) float v2f;
typedef __attribute__((ext_vector_type(8))) float v8f;

// ---------------------------------------------------------------------------
// Software grid barrier across G_BLOCKS persistent blocks.
// Deterministic w.r.t. data: only ordering of arrival differs.
// ---------------------------------------------------------------------------
__device__ __forceinline__ void grid_barrier(int* cnt, volatile int* gen, int nblk) {
    __syncthreads();
    if (threadIdx.x == 0) {
        int g = *gen;                      // read generation BEFORE signaling arrival
        __threadfence();
        if (atomicAdd(cnt, 1) == nblk - 1) {
            *cnt = 0;
            __threadfence();
            *gen = g + 1;                  // release everyone
        } else {
            while (*gen == g) { __builtin_amdgcn_s_sleep(1); }
        }
        __threadfence();
    }
    __syncthreads();
}

__global__ void pl_init_barrier(int* bar) {
    if (threadIdx.x == 0) { bar[0] = 0; bar[1] = 0; }
}

// ---------------------------------------------------------------------------
// Kernel 1: dense, batchable part.  base[t][j] = sum_k X[t][k] * W[k][j] + b.
// 2048x512x512 f32 GEMM on the matrix units: V_WMMA_F32_16X16X4_F32.
// One wave32 per 16x16 output tile, K-loop in steps of 4.
// A 16x4 f32 layout: lanes 0-15 hold M=0..15 {K=k0,k0+1}, lanes 16-31 {K=k0+2,k0+3}.
// B 4x16 mirrors the K split across lane halves.
// ---------------------------------------------------------------------------
__global__ __launch_bounds__(256) void pl_gemm_base(const float* __restrict__ X,
                                                    const float* __restrict__ W,
                                                    const float* __restrict__ B,
                                                    float* __restrict__ base) {
    const int wave = threadIdx.x >> 5;                 // 0..7
    const int lane = threadIdx.x & 31;
    const int tile = blockIdx.x * 8 + wave;            // 0..4095
    const int tm   = tile >> 5;                        // 0..127  (T/16)
    const int tn   = tile & 31;                        // 0..31   (NF/16)
    const int m0   = tm * 16;
    const int n0   = tn * 16;
    const int half = lane >> 4;                        // 0/1: K-half selector
    const int ml   = lane & 15;                        // row (A) / col (B,C)

    v8f acc = {};
#if __has_builtin(__builtin_amdgcn_wmma_f32_16x16x4_f32)
    for (int k0 = 0; k0 < NF; k0 += 4) {
        const int ka = k0 + 2 * half;
        v2f a = *(const v2f*)(X + (m0 + ml) * NF + ka);          // X[m][ka], X[m][ka+1]
        v2f b;
        b.x = W[ka * NF + n0 + ml];                              // W[ka  ][n]
        b.y = W[(ka + 1) * NF + n0 + ml];                        // W[ka+1][n]
        acc = __builtin_amdgcn_wmma_f32_16x16x4_f32(
            /*neg_a=*/false, a, /*neg_b=*/false, b,
            /*c_mod=*/(short)0, acc, /*reuse_a=*/false, /*reuse_b=*/false);
    }
#else
    // VALU fallback (same tile ownership as the WMMA C/D layout)
    for (int v = 0; v < 8; ++v) {
        const int m = v + 8 * half;
        float s = 0.f;
        for (int k = 0; k < NF; ++k)
            s = fmaf(X[(m0 + m) * NF + k], W[k * NF + n0 + ml], s);
        acc[v] = s;
    }
#endif
    const float bias = B[0];
    // C/D layout: VGPR v, lanes 0-15 -> M=v, lanes 16-31 -> M=v+8; N = ml.
#pragma unroll
    for (int v = 0; v < 8; ++v) {
        const int m = v + 8 * half;
        base[(m0 + m) * NF + n0 + ml] = acc[v] + bias;
    }
}

// ---------------------------------------------------------------------------
// Kernel 2: persistent sequential loop.  32 blocks x 512 threads.
// Block g owns trace rows [16g, 16g+16); thread j owns column j of that slab.
// trace/alpha/eta live in registers for the whole 2048-step loop.
// Per step: sum partials -> logits(+WMMA base) -> softmax -> trace update
// -> next-step partial, ONE grid barrier.
// ---------------------------------------------------------------------------
__global__ __launch_bounds__(512) void pl_loop(const float* __restrict__ X,
                                               const float* __restrict__ alpha,
                                               const float* __restrict__ eta,
                                               const float* __restrict__ base,
                                               float* __restrict__ out,
                                               float* __restrict__ partial,
                                               int* bar) {
    const int g  = blockIdx.x;
    const int j  = threadIdx.x;          // output column, 0..511
    const int i0 = g * ROWS;
    int* cnt = bar;
    volatile int* gen = (volatile int*)(bar + 1);

    float al[ROWS], et[ROWS], tr[ROWS], xc[ROWS];
#pragma unroll
    for (int r = 0; r < ROWS; ++r) {
        al[r] = alpha[(i0 + r) * NF + j];
        et[r] = eta[(i0 + r) * NF + j];
        tr[r] = 0.f;                      // trace starts at zero
    }
#pragma unroll
    for (int r = 0; r < ROWS; ++r)
        xc[r] = X[i0 + r];                // x_0 slab (uniform address -> scalar loads)

    // step-0 correction partial is zero (trace == 0)
    partial[0 * G_BLOCKS * NF + g * NF + j] = 0.f;
    grid_barrier(cnt, gen, G_BLOCKS);

    __shared__ float red[16];

    for (int t = 0; t < T_STEPS; ++t) {
        // ---- logits: WMMA base + sum of per-block correction partials ----
        float logit = base[t * NF + j];
        const float* pb = partial + (t & 1) * G_BLOCKS * NF;
        float csum = 0.f;
#pragma unroll
        for (int q = 0; q < G_BLOCKS; ++q)     // fixed order -> deterministic
            csum += pb[q * NF + j];
        logit += csum;

        // ---- softmax over 512 columns (16 waves) ----
        float m = logit;
#pragma unroll
        for (int d = 16; d >= 1; d >>= 1)
            m = fmaxf(m, __shfl_xor(m, d, 32));
        if ((j & 31) == 0) red[j >> 5] = m;
        __syncthreads();
        m = red[0];
#pragma unroll
        for (int q = 1; q < 16; ++q) m = fmaxf(m, red[q]);
        __syncthreads();                       // red[] reuse

        float e = __expf(logit - m);
        float s = e;
#pragma unroll
        for (int d = 16; d >= 1; d >>= 1)
            s += __shfl_xor(s, d, 32);
        if ((j & 31) == 0) red[j >> 5] = s;
        __syncthreads();
        s = 0.f;
#pragma unroll
        for (int q = 0; q < 16; ++q) s += red[q];
        const float y = e / s;

        if (g == 0) out[t * NF + j] = y;       // one block writes the row

        // ---- next-step x slab (uniform -> scalar loads) ----
        float xn[ROWS];
        if (t + 1 < T_STEPS) {
#pragma unroll
            for (int r = 0; r < ROWS; ++r) xn[r] = X[(t + 1) * NF + i0 + r];
        } else {
#pragma unroll
            for (int r = 0; r < ROWS; ++r) xn[r] = 0.f;
        }

        // ---- fused trace update + next-step correction partial ----
        float p = 0.f;
#pragma unroll
        for (int r = 0; r < ROWS; ++r) {
            const float xy = xc[r] * y;
            float nt = fmaf(et[r], xy - tr[r], tr[r]);   // (1-eta)*tr + eta*x*y
            nt = fminf(1.f, fmaxf(-1.f, nt));
            tr[r] = nt;
            p = fmaf(xn[r], al[r] * nt, p);              // x_{t+1} . (alpha o trace_{t+1})
        }
        partial[((t + 1) & 1) * G_BLOCKS * NF + g * NF + j] = p;
#pragma unroll
        for (int r = 0; r < ROWS; ++r) xc[r] = xn[r];

        grid_barrier(cnt, gen, G_BLOCKS);
        __syncthreads();                       // red[] reuse across steps
    }
}

// ---------------------------------------------------------------------------
extern "C" void kernel_launch(void* const* d_in, const int* in_sizes, int n_in,
                              void* d_out, int out_size, void* d_ws, size_t ws_size,
                              hipStream_t stream) {
    const float* x     = (const float*)d_in[0];   // [2048,512]
    const float* w     = (const float*)d_in[1];   // [512,512]
    const float* b     = (const float*)d_in[2];   // [1]
    const float* alpha = (const float*)d_in[3];   // [512,512]
    const float* eta   = (const float*)d_in[4];   // [512,512]
    // d_in[5] = is_training: trace starts at zero either way.
    float* out = (float*)d_out;                   // [2048,512]

    char*  ws      = (char*)d_ws;
    int*   bar     = (int*)ws;                                        // cnt, gen
    float* partial = (float*)(ws + 256);                              // [2][32][512]
    float* base    = (float*)(ws + 256 + 2 * G_BLOCKS * NF * sizeof(float)); // [2048][512]

    hipLaunchKernelGGL(pl_init_barrier, dim3(1), dim3(64), 0, stream, bar);

    const int tiles = (T_STEPS / 16) * (NF / 16);   // 4096 waves
    hipLaunchKernelGGL(pl_gemm_base, dim3(tiles / 8), dim3(256), 0, stream,
                       x, w, b, base);

    hipLaunchKernelGGL(pl_loop, dim3(G_BLOCKS), dim3(NF), 0, stream,
                       x, alpha, eta, base, out, partial, bar);
}